// ConvergedInhibition_29128468201923
// MI455X (gfx1250) — compile-verified
//
#include <hip/hip_runtime.h>
#include <math.h>

#define C_CH 63
#define HW   12544           // 112*112
#define NBATCH 64
#define GROUPS_PER_WAVE 8
#define WAVES_PER_BLOCK 8

typedef __attribute__((ext_vector_type(2))) float v2f;
typedef __attribute__((ext_vector_type(8))) float v8f;

// ---------------------------------------------------------------------------
// Kernel A: build the 64x64 zero-padded circulant deconvolution matrix M from
// the inhibition filter + kronecker delta.  M[c][j] = h[(c-j) mod 63] where
// h = Re(ifft(1/fft(delta - filter))).  Tiny: 1 block, 64 threads.
// ---------------------------------------------------------------------------
__global__ __launch_bounds__(64)
void ConvInhib_build_matrix(const float* __restrict__ filt,
                            const float* __restrict__ delta,
                            float* __restrict__ M)
{
    __shared__ float fkr[C_CH], fki[C_CH], h[C_CH];
    const int t = threadIdx.x;
    const double TWO_PI = 6.283185307179586476925286766559;

    if (t < C_CH) {
        double re = 0.0, im = 0.0;
        for (int n = 0; n < C_CH; ++n) {
            double g  = (double)delta[n] - (double)filt[n];
            double th = -TWO_PI * (double)(t * n) / (double)C_CH;
            re += g * cos(th);
            im += g * sin(th);
        }
        fkr[t] = (float)re;
        fki[t] = (float)im;
    }
    __syncthreads();

    if (t < C_CH) {
        double acc = 0.0;
        for (int m = 0; m < C_CH; ++m) {
            double a   = (double)fkr[m], b = (double)fki[m];
            double inv = 1.0 / (a * a + b * b);
            double ir  =  a * inv;          // Re(1/fk)
            double ii  = -b * inv;          // Im(1/fk)
            double th  = TWO_PI * (double)(m * t) / (double)C_CH;
            acc += ir * cos(th) - ii * sin(th);
        }
        h[t] = (float)(acc / (double)C_CH);
    }
    __syncthreads();

    // Thread t writes row t of the zero-padded 64x64 matrix (row-major).
    for (int j = 0; j < 64; ++j) {
        float v = 0.0f;
        if (t < C_CH && j < C_CH) v = h[(t - j + C_CH) % C_CH];
        M[t * 64 + j] = v;
    }
}

// ---------------------------------------------------------------------------
// Kernel B: Y = M x X as a streaming WMMA GEMM.
// Each wave: 16-pixel group, K padded to 64, 4 row-tiles of 16 output
// channels -> 64 x v_wmma_f32_16x16x4_f32 per group.  M lives in 128 VGPRs
// per wave (loaded once, amortized over GROUPS_PER_WAVE groups).
// A layout (16x4 f32): lanes 0-15 = M rows, K = 2*khalf + v per VGPR v.
// B layout (4x16 f32): lanes 0-15 = pixels, K rows striped the same way.
// ---------------------------------------------------------------------------
__global__ __launch_bounds__(256)
void ConvInhib_deconv_gemm(const float* __restrict__ act,
                           const float* __restrict__ M,
                           float* __restrict__ out)
{
    const int lane  = threadIdx.x & 31;
    const int wave  = threadIdx.x >> 5;
    const int lrow  = lane & 15;   // A: output row within tile / B,D: pixel
    const int khalf = lane >> 4;   // selects K pair (A/B) or row+8 (D)

    // Whole 64x64 M in registers, WMMA A-operand layout, v2f = (K, K+1).
    v2f A[4][16];
#pragma unroll
    for (int r = 0; r < 4; ++r)
#pragma unroll
        for (int s = 0; s < 16; ++s)
            A[r][s] = *(const v2f*)(M + (16 * r + lrow) * 64 + 4 * s + 2 * khalf);

    const int gBase = blockIdx.x * (WAVES_PER_BLOCK * GROUPS_PER_WAVE)
                    + wave * GROUPS_PER_WAVE;

    for (int i = 0; i < GROUPS_PER_WAVE; ++i) {
        const int g   = gBase + i;
        const int n   = g / (HW / 16);          // batch image (HW%16==0: never straddles)
        const int hw0 = (g % (HW / 16)) * 16;   // first pixel of the group
        const float* src = act + (size_t)n * C_CH * HW + hw0 + lrow;
        float*       dst = out + (size_t)n * C_CH * HW + hw0 + lrow;

        // B operands: 16 K-steps; streamed once -> non-temporal loads.
        // Channel 63 is padding: clamp address (weight column 63 is zero).
        v2f B[16];
#pragma unroll
        for (int s = 0; s < 16; ++s) {
            int c0 = 4 * s + 2 * khalf;
            int c1 = c0 + 1;
            if (c0 > C_CH - 1) c0 = C_CH - 1;
            if (c1 > C_CH - 1) c1 = C_CH - 1;
            B[s].x = __builtin_nontemporal_load(src + (size_t)c0 * HW);
            B[s].y = __builtin_nontemporal_load(src + (size_t)c1 * HW);
        }

        v8f acc0 = {}, acc1 = {}, acc2 = {}, acc3 = {};
#pragma unroll
        for (int s = 0; s < 16; ++s) {
            acc0 = __builtin_amdgcn_wmma_f32_16x16x4_f32(false, A[0][s], false, B[s],
                                                         (short)0, acc0, false, false);
            acc1 = __builtin_amdgcn_wmma_f32_16x16x4_f32(false, A[1][s], false, B[s],
                                                         (short)0, acc1, false, false);
            acc2 = __builtin_amdgcn_wmma_f32_16x16x4_f32(false, A[2][s], false, B[s],
                                                         (short)0, acc2, false, false);
            acc3 = __builtin_amdgcn_wmma_f32_16x16x4_f32(false, A[3][s], false, B[s],
                                                         (short)0, acc3, false, false);
        }

        // D layout: VGPR j -> row (16r + 8*khalf + j), lane = pixel.
#pragma unroll
        for (int j = 0; j < 8; ++j) {
            const int row0 =      8 * khalf + j;
            const int row1 = 16 + 8 * khalf + j;
            const int row2 = 32 + 8 * khalf + j;
            __builtin_nontemporal_store(acc0[j], dst + (size_t)row0 * HW);
            __builtin_nontemporal_store(acc1[j], dst + (size_t)row1 * HW);
            __builtin_nontemporal_store(acc2[j], dst + (size_t)row2 * HW);
        }
#pragma unroll
        for (int j = 0; j < 8; ++j) {
            const int row3 = 48 + 8 * khalf + j;
            if (row3 < C_CH)    // skip padded output row 63
                __builtin_nontemporal_store(acc3[j], dst + (size_t)row3 * HW);
        }
    }
}

extern "C" void kernel_launch(void* const* d_in, const int* in_sizes, int n_in,
                              void* d_out, int out_size, void* d_ws, size_t ws_size,
                              hipStream_t stream) {
    (void)in_sizes; (void)n_in; (void)out_size; (void)ws_size;
    const float* act   = (const float*)d_in[0];  // (64,63,112,112) f32
    const float* filt  = (const float*)d_in[1];  // (1,1,1,63) f32
    const float* delta = (const float*)d_in[2];  // (1,1,1,63) f32
    float* M   = (float*)d_ws;                   // 64*64 f32 scratch
    float* out = (float*)d_out;

    ConvInhib_build_matrix<<<1, 64, 0, stream>>>(filt, delta, M);

    const int groups = (NBATCH * HW) / 16;                           // 50176
    const int blocks = groups / (WAVES_PER_BLOCK * GROUPS_PER_WAVE); // 784
    ConvInhib_deconv_gemm<<<blocks, 256, 0, stream>>>(act, M, out);
}